// CostVolume_30176440222488
// MI455X (gfx1250) — compile-verified
//
#include <hip/hip_runtime.h>

typedef float v4f __attribute__((ext_vector_type(4)));
typedef int   v4i __attribute__((ext_vector_type(4)));

// Address-space-qualified vector pointee types for the async-to-LDS builtin:
// prototype (per round-1 diagnostic) takes AS1/AS3 pointers to 4xi32.
typedef __attribute__((address_space(1))) v4i as1_v4i;
typedef __attribute__((address_space(3))) v4i as3_v4i;

#ifndef __has_builtin
#define __has_builtin(x) 0
#endif

__device__ __forceinline__ float clamp1000(float x) {
    return __builtin_fminf(__builtin_fmaxf(x, -1000.0f), 1000.0f);
}

// One block per (c,h) row pair: stage left/right rows (W floats each) into LDS
// via async global->LDS b128 copies (ASYNCcnt path), then emit D x (W/4)
// non-temporal float4 stores. Store-bandwidth bound: ~400 MiB output.
template <int W, int D, int H>
__global__ __launch_bounds__(256) void costvol_kernel(
    const float* __restrict__ left,
    const float* __restrict__ right,
    float* __restrict__ out)
{
    __shared__ float lrow[W];
    __shared__ float rrow[W];

    constexpr int WV = W / 4;           // float4 per row (80)
    const int t  = threadIdx.x;
    const int ch = blockIdx.x;          // ch = c*H + h
    const int c  = ch / H;              // constexpr H -> mul/shift, no div seq
    const int h  = ch - c * H;

    const float* lp = left  + (size_t)ch * W;
    const float* rp = right + (size_t)ch * W;

#if defined(__gfx1250__) && __has_builtin(__builtin_amdgcn_global_load_async_to_lds_b128)
    // Async DMA of both rows into LDS: 2*WV = 160 lanes issue one b128 each.
    if (t < 2 * WV) {
        const float* src = (t < WV) ? (lp + 4 * t) : (rp + 4 * (t - WV));
        float*       dst = (t < WV) ? (lrow + 4 * t) : (rrow + 4 * (t - WV));
        __builtin_amdgcn_global_load_async_to_lds_b128(
            (as1_v4i*)src, (as3_v4i*)dst, /*offset=*/0, /*cpol=*/0);
    }
  #if __has_builtin(__builtin_amdgcn_s_wait_asynccnt)
    __builtin_amdgcn_s_wait_asynccnt(0);
  #else
    asm volatile("s_wait_asynccnt 0" ::: "memory");
  #endif
    __syncthreads();
#else
    for (int i = t; i < W; i += 256) {
        lrow[i] = lp[i];
        rrow[i] = rp[i];
    }
    __syncthreads();
#endif

    // out[((c*D + d)*H + h)*W + w]
    float* obase = out + ((size_t)c * D * H + h) * W;
    constexpr size_t dstride = (size_t)H * W;

    constexpr int NV = WV * D;          // 5120 vec4 stores per block (20/thread)
    #pragma unroll 4
    for (int idx = t; idx < NV; idx += 256) {
        const int d  = idx / WV;        // constexpr WV -> mul/shift
        const int w4 = idx - d * WV;
        const int w0 = 4 * w4;
        v4f v;
        if (w0 >= d) {
            // all 4 lanes valid: straight LDS reads
            const int s = w0 - d;
            v[0] = clamp1000(lrow[w0 + 0] * rrow[s + 0]);
            v[1] = clamp1000(lrow[w0 + 1] * rrow[s + 1]);
            v[2] = clamp1000(lrow[w0 + 2] * rrow[s + 2]);
            v[3] = clamp1000(lrow[w0 + 3] * rrow[s + 3]);
        } else {
            #pragma unroll
            for (int j = 0; j < 4; ++j) {
                const int w = w0 + j;
                v[j] = (w >= d) ? clamp1000(lrow[w] * rrow[w - d]) : 0.0f;
            }
        }
        // Write-once 400 MiB stream: non-temporal so it doesn't thrash the
        // L2 lines holding the (tiny) inputs.
        __builtin_nontemporal_store(v, (v4f*)(obase + (size_t)d * dstride + w0));
    }
}

extern "C" void kernel_launch(void* const* d_in, const int* in_sizes, int n_in,
                              void* d_out, int out_size, void* d_ws, size_t ws_size,
                              hipStream_t stream) {
    const float* left  = (const float*)d_in[0];
    const float* right = (const float*)d_in[1];
    float*       out   = (float*)d_out;

    constexpr int C = 32, H = 160, W = 320, D = 64;  // B = 1
    dim3 grid(C * H);
    dim3 block(256);
    costvol_kernel<W, D, H><<<grid, block, 0, stream>>>(left, right, out);
}